// GCN_test_59270548684944
// MI455X (gfx1250) — compile-verified
//
#include <hip/hip_runtime.h>
#include <hip/hip_bf16.h>

typedef __attribute__((ext_vector_type(2))) float v2f;
typedef __attribute__((ext_vector_type(8))) float v8f;

#define N_NODES 1395
#define MPAD    1408          // 88 * 16 : node dim padded to 16 for WMMA tiles
#define C3      192
#define TILES   ((MPAD/16)*(C3/16))   // 88*12 = 1056

// ---------------- utility kernels ----------------
__global__ void k_zero(float* __restrict__ p, int n) {
  int i = blockIdx.x * 256 + threadIdx.x;
  if (i < n) p[i] = 0.f;
}

__global__ void k_copy(float* __restrict__ d, const float* __restrict__ s, int n) {
  int i = blockIdx.x * 256 + threadIdx.x;
  if (i < n) d[i] = s[i];
}

// x0 = x with first 3 feature columns zeroed
__global__ void k_prep(const float* __restrict__ x, float* __restrict__ x0, int n) {
  int i = blockIdx.x * 256 + threadIdx.x;
  if (i < n * 6) {
    int c = i % 6;
    x0[i] = (c < 3) ? 0.f : x[i];
  }
}

// degree count over dst (deg pre-zeroed); self-loop folded into k_dinv
__global__ void k_deg(const int* __restrict__ dst, float* __restrict__ deg, int E) {
  int e = blockIdx.x * 256 + threadIdx.x;
  if (e < E) atomicAdd(&deg[dst[e]], 1.f);
}

__global__ void k_dinv(float* __restrict__ deg, int n) {
  int i = blockIdx.x * 256 + threadIdx.x;
  if (i < n) deg[i] = rsqrtf(deg[i] + 1.f);   // +1 = self loop
}

// dense Y[M,Nc] = X[M,K] @ W[K,Nc]  (+ optional bias)
__global__ void k_lin(const float* __restrict__ X, const float* __restrict__ W,
                      const float* __restrict__ bias, float* __restrict__ Y,
                      int M, int K, int Nc) {
  int idx = blockIdx.x * 256 + threadIdx.x;
  if (idx >= M * Nc) return;
  int c = idx % Nc, m = idx / Nc;
  float acc = bias ? bias[c] : 0.f;
  for (int k = 0; k < K; ++k) acc += X[m * K + k] * W[k * Nc + c];
  Y[idx] = acc;
}

// agg[dst] += dinv[src]*dinv[dst] * h[src]   over E real edges + n self loops
__global__ void k_scatter(const float* __restrict__ h, const int* __restrict__ src,
                          const int* __restrict__ dst, const float* __restrict__ dinv,
                          float* __restrict__ agg, int E, int n, int C) {
  long idx = (long)blockIdx.x * 256 + threadIdx.x;
  long total = (long)(E + n) * C;
  if (idx >= total) return;
  int c = (int)(idx % C);
  int e = (int)(idx / C);
  int s, d;
  if (e < E) { s = src[e]; d = dst[e]; } else { s = d = e - E; }
  float coef = dinv[s] * dinv[d];
  atomicAdd(&agg[d * C + c], coef * h[s * C + c]);
}

// in-place a = LeakyReLU(a + bias, 0.2)
__global__ void k_biasleaky(float* __restrict__ a, const float* __restrict__ b,
                            int C, int total) {
  int i = blockIdx.x * 256 + threadIdx.x;
  if (i < total) {
    float v = a[i] + b[i % C];
    a[i] = (v > 0.f) ? v : 0.2f * v;
  }
}

// ---------------- WMMA 192x192 per-node linear (qconv2/qconv3) ----------------
// O[MPAD,192] = H[MPAD,192] @ W[192,192] + b  using V_WMMA_F32_16X16X4_F32.
// One wave per 16x16 output tile; K marched in steps of 4 (48 WMMAs/wave).
// H is padded to MPAD rows with zeros so EXEC stays all-1s (WMMA requirement).
__global__ void k_qconv_wmma(const float* __restrict__ H, const float* __restrict__ W,
                             const float* __restrict__ b, float* __restrict__ O) {
  int wave = threadIdx.x >> 5;                 // 8 waves / block
  int lane = threadIdx.x & 31;
  int tile = blockIdx.x * 8 + wave;            // exactly TILES tiles launched
  int tm = tile / (C3 / 16);
  int tn = tile % (C3 / 16);
  int row0 = tm * 16;
  int col0 = tn * 16;

  int m     = lane & 15;                       // A: lanes 0-15 hold M=0..15
  int khalf = (lane >> 4) << 1;                // VGPR pair covers K 0/1 (lanes<16) or 2/3
  const float* arow = H + (size_t)(row0 + m) * C3;
  int ncol = col0 + m;                         // B: N striped across lanes (mirror of A)

  v8f c = {};
  for (int k = 0; k < C3; k += 4) {
    v2f a, bb;
    a.x  = arow[k + khalf];
    a.y  = arow[k + khalf + 1];
    bb.x = W[(size_t)(k + khalf)     * C3 + ncol];
    bb.y = W[(size_t)(k + khalf + 1) * C3 + ncol];
    c = __builtin_amdgcn_wmma_f32_16x16x4_f32(
        /*neg_a=*/false, a, /*neg_b=*/false, bb,
        /*c_mod=*/(short)0, c, /*reuse_a=*/false, /*reuse_b=*/false);
  }

  // D layout: VGPR r -> M = row0 + r (+8 for lanes 16-31), N = col0 + (lane&15)
  int nout  = col0 + (lane & 15);
  int mbase = row0 + ((lane >> 4) << 3);
  float bias = b[nout];
#pragma unroll
  for (int r = 0; r < 8; ++r)
    O[(size_t)(mbase + r) * C3 + nout] = c[r] + bias;
}

// ---------------- fc1: the 1.07 GB streaming GEMV (HBM-bound) ----------------
// o[1024] += sum_i v[i] * W[i,1024].  Split-K over blocks; each block streams a
// 128-row slab of W with float4 coalesced reads, then 4 atomics/thread.
__global__ void k_fc1(const float* __restrict__ v, const float* __restrict__ W,
                      float* __restrict__ o, int rows, int rowsPerBlock) {
  int r0 = blockIdx.x * rowsPerBlock;
  int r1 = r0 + rowsPerBlock; if (r1 > rows) r1 = rows;
  int t = threadIdx.x;                          // 256 threads * float4 = 1024 cols
  const float4* W4 = (const float4*)W;
  float4 acc = make_float4(0.f, 0.f, 0.f, 0.f);
  for (int i = r0; i < r1; ++i) {
    float vi = v[i];
    float4 w = W4[(size_t)i * 256 + t];
    acc.x += vi * w.x; acc.y += vi * w.y; acc.z += vi * w.z; acc.w += vi * w.w;
  }
  atomicAdd(&o[4 * t + 0], acc.x);
  atomicAdd(&o[4 * t + 1], acc.y);
  atomicAdd(&o[4 * t + 2], acc.z);
  atomicAdd(&o[4 * t + 3], acc.w);
}

// fc2: o2[1024] += split-K GEMV over 1024x1024
__global__ void k_fc2(const float* __restrict__ x, const float* __restrict__ W,
                      float* __restrict__ o) {
  int jb = blockIdx.x & 3;                      // 4 column blocks of 256
  int kb = blockIdx.x >> 2;                     // 16 K-slabs of 64
  int j = jb * 256 + threadIdx.x;
  float acc = 0.f;
  int i0 = kb * 64;
  for (int i = i0; i < i0 + 64; ++i) acc += x[i] * W[(size_t)i * 1024 + j];
  atomicAdd(&o[j], acc);
}

// fc3 (1024->64) + quaternion group-of-4 L2 normalize; single 64-thread block
__global__ void k_fc3norm(const float* __restrict__ x, const float* __restrict__ W,
                          const float* __restrict__ b, float* __restrict__ out) {
  int j = threadIdx.x;                          // 0..63
  float acc = b[j];
  for (int i = 0; i < 1024; ++i) acc += x[i] * W[(size_t)i * 64 + j];
  __shared__ float s[64];
  s[j] = acc * acc;
  __syncthreads();
  int g = j & ~3;
  float mag = sqrtf(s[g] + s[g + 1] + s[g + 2] + s[g + 3]);
  out[j] = acc / mag;
}

// ---------------- launch ----------------
extern "C" void kernel_launch(void* const* d_in, const int* in_sizes, int n_in,
                              void* d_out, int out_size, void* d_ws, size_t ws_size,
                              hipStream_t stream) {
  const float* x    = (const float*)d_in[0];
  const int*   ei   = (const int*)  d_in[1];
  const float* w1   = (const float*)d_in[2];   const float* b1  = (const float*)d_in[3];
  const float* w2   = (const float*)d_in[4];   const float* b2  = (const float*)d_in[5];
  const float* qw1  = (const float*)d_in[6];   const float* qb1 = (const float*)d_in[7];
  const float* qw2  = (const float*)d_in[8];   const float* qb2 = (const float*)d_in[9];
  const float* qw3  = (const float*)d_in[10];  const float* qb3 = (const float*)d_in[11];
  const float* fw1  = (const float*)d_in[12];  const float* fb1 = (const float*)d_in[13];
  const float* fw2  = (const float*)d_in[14];  const float* fb2 = (const float*)d_in[15];
  const float* fw3  = (const float*)d_in[16];  const float* fb3 = (const float*)d_in[17];
  float* out = (float*)d_out;

  const int n = in_sizes[0] / 6;               // 1395
  const int E = in_sizes[1] / 2;               // 44640
  const int* src = ei;
  const int* dst = ei + E;

  // workspace layout (floats); zero-initialized region first
  float* ws = (float*)d_ws;
  size_t off = 0;
  float* deg  = ws + off; off += n;            // deg -> dinv (in place)
  float* agg1 = ws + off; off += (size_t)n * 32;
  float* agg2 = ws + off; off += (size_t)n * 30;
  float* H3   = ws + off; off += (size_t)MPAD * C3;   // also reused as H5
  float* H4   = ws + off; off += (size_t)MPAD * C3;
  size_t zeroN = off;                           // everything above must start at 0
  float* x0   = ws + off; off += (size_t)n * 6;
  float* h1   = ws + off; off += (size_t)n * 32;
  float* h2   = ws + off; off += (size_t)n * 30;
  float* o1   = ws + off; off += 1024;
  float* o2   = ws + off; off += 1024;
  float* H5   = H3;                             // alias: H3 is dead after qconv2

  auto blks = [](long total) { return (int)((total + 255) / 256); };

  // init accumulators every call (harness does not re-zero ws between replays)
  k_zero<<<blks((long)zeroN), 256, 0, stream>>>(ws, (int)zeroN);
  k_copy<<<4, 256, 0, stream>>>(o1, fb1, 1024);  // seed with bias for split-K atomics
  k_copy<<<4, 256, 0, stream>>>(o2, fb2, 1024);

  // preprocessing + GCN normalization
  k_prep<<<blks((long)n * 6), 256, 0, stream>>>(x, x0, n);
  k_deg <<<blks(E), 256, 0, stream>>>(dst, deg, E);
  k_dinv<<<blks(n), 256, 0, stream>>>(deg, n);

  // GCN layer 1: 6 -> 32
  k_lin<<<blks((long)n * 32), 256, 0, stream>>>(x0, w1, nullptr, h1, n, 6, 32);
  k_scatter<<<blks((long)(E + n) * 32), 256, 0, stream>>>(h1, src, dst, deg, agg1, E, n, 32);
  k_biasleaky<<<blks((long)n * 32), 256, 0, stream>>>(agg1, b1, 32, n * 32);

  // GCN layer 2: 32 -> 30
  k_lin<<<blks((long)n * 30), 256, 0, stream>>>(agg1, w2, nullptr, h2, n, 32, 30);
  k_scatter<<<blks((long)(E + n) * 30), 256, 0, stream>>>(h2, src, dst, deg, agg2, E, n, 30);
  k_biasleaky<<<blks((long)n * 30), 256, 0, stream>>>(agg2, b2, 30, n * 30);

  // qconv1: 30 -> 192 (pad rows of H3 stay zero from init)
  k_lin<<<blks((long)n * C3), 256, 0, stream>>>(agg2, qw1, qb1, H3, n, 30, C3);

  // qconv2 / qconv3: 192x192 GEMMs on the WMMA f32 path
  k_qconv_wmma<<<TILES / 8, 256, 0, stream>>>(H3, qw2, qb2, H4);
  k_qconv_wmma<<<TILES / 8, 256, 0, stream>>>(H4, qw3, qb3, H5);

  // fc1: the HBM-roofline stage — stream 1.07 GB of fw1
  const int rows = n * C3;                      // 267840 (pad rows excluded)
  const int rpb  = 128;
  k_fc1<<<(rows + rpb - 1) / rpb, 256, 0, stream>>>(H5, fw1, o1, rows, rpb);

  // fc2 + fc3 + group-of-4 normalize
  k_fc2<<<64, 256, 0, stream>>>(o1, fw2, o2);
  k_fc3norm<<<1, 64, 0, stream>>>(o2, fw3, fb3, out);
}